// LIFOutSpike_75299366633688
// MI455X (gfx1250) — compile-verified
//
#include <hip/hip_runtime.h>
#include <math.h>

// LIF spiking recurrence, STEPS=4, memory-bound streaming kernel for MI455X.
// 768 MB total traffic @ 23.3 TB/s -> ~33us floor. No GEMM structure -> no WMMA.
// CDNA5 paths used: wave32 sizing, global_{load,store}_b128 with non-temporal
// TH cache hints (stream 768MB past the 192MB L2 without pollution).

typedef float v4f __attribute__((ext_vector_type(4)));

#define TAU_F 0.25f

// One (b,n) site: 4-step LIF scan over a float4 (time is the contiguous axis).
// Evaluation order matches the reference: u = ((TAU*u)*reset + x_t) + l_t,
// forced with IEEE intrinsics so fp-contract can't perturb spike decisions.
__device__ __forceinline__ v4f lif_row(v4f xv, v4f lv, float thre) {
    v4f r;

    // t = 0: u0 = 0, o0 = 0  =>  u = x + l regardless of reset.
    float u = __fadd_rn(xv.x, lv.x);
    float o = (u > thre) ? u : 0.0f;
    r.x = o;

    // t = 1
    float reset = (o > thre) ? 0.0f : 1.0f;
    u = __fadd_rn(__fadd_rn(__fmul_rn(__fmul_rn(TAU_F, u), reset), xv.y), lv.y);
    o = (u > thre) ? u : 0.0f;
    r.y = o;

    // t = 2
    reset = (o > thre) ? 0.0f : 1.0f;
    u = __fadd_rn(__fadd_rn(__fmul_rn(__fmul_rn(TAU_F, u), reset), xv.z), lv.z);
    o = (u > thre) ? u : 0.0f;
    r.z = o;

    // t = 3
    reset = (o > thre) ? 0.0f : 1.0f;
    u = __fadd_rn(__fadd_rn(__fmul_rn(__fmul_rn(TAU_F, u), reset), xv.w), lv.w);
    o = (u > thre) ? u : 0.0f;
    r.w = o;

    return r;
}

__global__ __launch_bounds__(256) void lif_outspike_kernel(
    const float* __restrict__ x,    // [B*N, 4]
    const float* __restrict__ lat,  // [B*N, 4]
    const float* __restrict__ w,    // scalar
    float* __restrict__ out,        // [B*N, 4]
    int n_rows) {
    const float thre = tanhf(w[0]);  // uniform across the wave; cost amortized
                                     // over 96 B of HBM traffic per thread.

    const v4f* __restrict__ xv = (const v4f*)x;
    const v4f* __restrict__ lv = (const v4f*)lat;
    v4f* __restrict__ ov = (v4f*)out;

    // 2 rows per thread; second row offset by blockDim.x so every b128 load in
    // a wave32 covers a contiguous 512B span (fully coalesced), and each thread
    // has 4 loads in flight before its first s_wait_loadcnt.
    const int base = blockIdx.x * (blockDim.x * 2) + threadIdx.x;
    const int r0 = base;
    const int r1 = base + blockDim.x;

    if (r1 < n_rows) {
        v4f x0 = __builtin_nontemporal_load(xv + r0);
        v4f l0 = __builtin_nontemporal_load(lv + r0);
        v4f x1 = __builtin_nontemporal_load(xv + r1);
        v4f l1 = __builtin_nontemporal_load(lv + r1);
        __builtin_nontemporal_store(lif_row(x0, l0, thre), ov + r0);
        __builtin_nontemporal_store(lif_row(x1, l1, thre), ov + r1);
    } else if (r0 < n_rows) {
        v4f x0 = __builtin_nontemporal_load(xv + r0);
        v4f l0 = __builtin_nontemporal_load(lv + r0);
        __builtin_nontemporal_store(lif_row(x0, l0, thre), ov + r0);
    }
}

extern "C" void kernel_launch(void* const* d_in, const int* in_sizes, int n_in,
                              void* d_out, int out_size, void* d_ws, size_t ws_size,
                              hipStream_t stream) {
    const float* x   = (const float*)d_in[0];  // [16, 1048576, 4] f32
    const float* lat = (const float*)d_in[1];  // [16, 1048576, 4] f32
    const float* w   = (const float*)d_in[2];  // scalar f32
    float* out = (float*)d_out;                // [16, 1048576, 4] f32

    const int steps  = 4;
    const int n_rows = in_sizes[0] / steps;    // 16,777,216 (b,n) sites

    const int threads = 256;                   // 8 wave32s per block
    const int rows_per_block = threads * 2;
    const int blocks = (n_rows + rows_per_block - 1) / rows_per_block;  // 32768

    lif_outspike_kernel<<<blocks, threads, 0, stream>>>(x, lat, w, out, n_rows);
}